// ContrastiveTopK_86569360818416
// MI455X (gfx1250) — compile-verified
//
#include <hip/hip_runtime.h>
#include <math.h>

// ---------------- problem constants (from reference) ----------------
#define V_TOK   50257
#define B_ROWS  512
#define KEEP    45232            // ceil(0.9 * 50257)
#define TOPK2   40
#define NBINS   2048
#define CAP     1536             // gather buffer per target bin
#define TA      1024             // threads per block, kernel A (32 waves of 32)
#define LN_ALPHA (-2.3025850929940455f)   // log(0.1)

__device__ __forceinline__ float neg_inf_f() { return __uint_as_float(0xFF800000u); }
__device__ __forceinline__ float pos_inf_f() { return __uint_as_float(0x7F800000u); }

// order-preserving float -> uint key (monotone increasing)
__device__ __forceinline__ unsigned fkey(float f) {
  unsigned u = __float_as_uint(f);
  return (u & 0x80000000u) ? ~u : (u | 0x80000000u);
}
__device__ __forceinline__ float fkey_inv(unsigned k) {
  unsigned u = (k & 0x80000000u) ? (k & 0x7FFFFFFFu) : ~k;
  return __uint_as_float(u);
}

// ================= Kernel A: per-row stats =================
// One block per (matrix,row). Row staged in LDS via CDNA5 async global->LDS
// (b128 bulk, phase-matched LDS placement so both sides are 16B aligned).
// Produces per row r (8 floats in ws):
//   exp: [0]=rowmax  [1]=Z (softmax denom over top-KEEP)  [2]=output threshold
//   ama: [4]=rowmax  [5]=Z                                [6]=keep threshold
__global__ void __launch_bounds__(TA) row_stats_kernel(
    const float* __restrict__ g_exp, const float* __restrict__ g_ama,
    float* __restrict__ stats)
{
  extern __shared__ float row[];               // V_TOK+4 floats (201 KB dynamic LDS)
  __shared__ unsigned hist[NBINS];
  __shared__ float    cand[2][CAP];
  __shared__ unsigned ccnt[2];
  __shared__ float    redF[32];
  __shared__ int      redI[32];
  __shared__ float    s_bcastF;
  __shared__ int      s_bcastI;
  __shared__ float    s_lo, s_hi;
  __shared__ int      s_bin[2];
  __shared__ int      s_rank[2];

  const int  tid   = threadIdx.x;
  const int  lane  = tid & 31;
  const int  wv    = tid >> 5;
  const int  bid   = blockIdx.x;
  const bool isExp = (bid < B_ROWS);
  const int  r     = isExp ? bid : (bid - B_ROWS);
  const float* __restrict__ src = (isExp ? g_exp : g_ama) + (size_t)r * V_TOK;

  // ---- async copy whole row: global -> LDS (ASYNCcnt path, gfx1250) ----
  // Phase-match the LDS image to the global row so b128 is aligned on BOTH sides:
  //   element i lives at LDS byte  lb + shiftB + 4*i,  with (lb+shiftB) == gb (mod 16).
  const unsigned gb = (unsigned)(size_t)src;
  const unsigned lb = (unsigned)(size_t)(&row[0]);
  const unsigned shiftB = (gb - lb) & 12u;       // 0,4,8,12 (both 4B aligned)
  const unsigned sh     = shiftB >> 2;           // element shift 0..3
  const unsigned ldsE0  = lb + shiftB;           // LDS byte addr of element 0

  const int i0 = (int)(((16u - (gb & 15u)) & 15u) >> 2); // first 16B-aligned element
  const int n4 = (V_TOK - i0) >> 2;                      // # of b128 chunks
  const int tail0 = i0 + (n4 << 2);

  if (tid < i0) {                                 // head (<=3 elements), b32
    const float* gp = src + tid;
    unsigned la = ldsE0 + (unsigned)tid * 4u;
    asm volatile("global_load_async_to_lds_b32 %0, %1, off"
                 :: "v"(la), "v"(gp) : "memory");
  }
  for (int c = tid; c < n4; c += TA) {            // bulk, b128 (16B aligned both sides)
    int i = i0 + (c << 2);
    const float* gp = src + i;
    unsigned la = ldsE0 + (unsigned)i * 4u;
    asm volatile("global_load_async_to_lds_b128 %0, %1, off"
                 :: "v"(la), "v"(gp) : "memory");
  }
  if (tail0 + tid < V_TOK) {                      // tail (<=3 elements), b32
    const float* gp = src + tail0 + tid;
    unsigned la = ldsE0 + (unsigned)(tail0 + tid) * 4u;
    asm volatile("global_load_async_to_lds_b32 %0, %1, off"
                 :: "v"(la), "v"(gp) : "memory");
  }
  asm volatile("s_wait_asynccnt 0" ::: "memory");
  __syncthreads();

  const float* rowp = &row[sh];                   // element i == rowp[i] (ds_load)

  // ---- pass 1: row min / max ----
  float vmax = neg_inf_f(), vmin = pos_inf_f();
  for (int i = tid; i < V_TOK; i += TA) {
    float x = rowp[i];
    vmax = fmaxf(vmax, x);
    vmin = fminf(vmin, x);
  }
  for (int o = 16; o; o >>= 1) {
    vmax = fmaxf(vmax, __shfl_xor(vmax, o, 32));
    vmin = fminf(vmin, __shfl_xor(vmin, o, 32));
  }
  if (lane == 0) redF[wv] = vmax;
  __syncthreads();
  if (wv == 0) {
    float t = redF[lane];
    for (int o = 16; o; o >>= 1) t = fmaxf(t, __shfl_xor(t, o, 32));
    if (lane == 0) s_hi = t;
  }
  __syncthreads();
  if (lane == 0) redF[wv] = vmin;
  __syncthreads();
  if (wv == 0) {
    float t = redF[lane];
    for (int o = 16; o; o >>= 1) t = fminf(t, __shfl_xor(t, o, 32));
    if (lane == 0) s_lo = t;
  }
  __syncthreads();
  const float m  = s_hi;
  const float lo = s_lo;
  const float range = m - lo;
  const float scale = (range > 0.f) ? ((float)NBINS / range) : 0.f;

  // ---- pass 2: linear histogram over [lo, hi] ----
  for (int i = tid; i < NBINS; i += TA) hist[i] = 0u;
  __syncthreads();
  for (int i = tid; i < V_TOK; i += TA) {
    float x = rowp[i];
    int b = (int)((x - lo) * scale);
    b = b < 0 ? 0 : (b >= NBINS ? NBINS - 1 : b);
    atomicAdd(&hist[b], 1u);
  }
  __syncthreads();

  // ---- locate bins holding rank KEEP (and rank 40 for exp) from the top ----
  if (tid == 0) {
    unsigned cum = 0;
    bool f0 = false, f1 = !isExp;
    s_bin[0] = 0; s_rank[0] = 1; s_bin[1] = 0; s_rank[1] = 1;
    for (int b = NBINS - 1; b >= 0; --b) {
      unsigned c = hist[b];
      if (!f1 && cum + c >= (unsigned)TOPK2) { s_bin[1] = b; s_rank[1] = TOPK2 - (int)cum; f1 = true; }
      if (!f0 && cum + c >= (unsigned)KEEP)  { s_bin[0] = b; s_rank[0] = KEEP  - (int)cum; f0 = true; }
      cum += c;
      if (f0 && f1) break;
    }
    ccnt[0] = 0u; ccnt[1] = 0u;
  }
  __syncthreads();

  // ---- pass 3: gather the (at most two) target bins into small buffers ----
  const int b0 = s_bin[0];
  const int b1 = isExp ? s_bin[1] : -1;
  for (int i = tid; i < V_TOK; i += TA) {
    float x = rowp[i];
    int b = (int)((x - lo) * scale);
    b = b < 0 ? 0 : (b >= NBINS ? NBINS - 1 : b);
    if (b == b0) { unsigned p = atomicAdd(&ccnt[0], 1u); if (p < CAP) cand[0][p] = x; }
    if (b == b1) { unsigned p = atomicAdd(&ccnt[1], 1u); if (p < CAP) cand[1][p] = x; }
  }
  __syncthreads();

  // ---- exact rank value inside each target bin: 32-step key bisection ----
  float thr[2]; thr[0] = 0.f; thr[1] = 0.f;
  const int nsel = isExp ? 2 : 1;
  for (int j = 0; j < nsel; ++j) {
    unsigned cc = ccnt[j];
    int mcnt = (int)(cc < (unsigned)CAP ? cc : (unsigned)CAP);
    int rr = s_rank[j];
    if (rr > mcnt) rr = mcnt;
    if (rr < 1)    rr = 1;
    unsigned ans = 0u;
    for (int bit = 31; bit >= 0; --bit) {
      unsigned trial = ans | (1u << bit);
      int c = 0;
      for (int t2 = tid; t2 < mcnt; t2 += TA)
        c += (fkey(cand[j][t2]) >= trial) ? 1 : 0;
      for (int o = 16; o; o >>= 1) c += __shfl_xor(c, o, 32);
      if (lane == 0) redI[wv] = c;
      __syncthreads();
      if (wv == 0) {
        int t = redI[lane];
        for (int o = 16; o; o >>= 1) t += __shfl_xor(t, o, 32);
        if (lane == 0) s_bcastI = t;
      }
      __syncthreads();
      if (s_bcastI >= rr) ans = trial;    // uniform decision
      __syncthreads();                    // protect s_bcastI vs next write
    }
    thr[j] = fkey_inv(ans);               // exact rr-th largest in that bin
  }

  // ---- pass 4: softmax denominator over kept entries (x >= tmain) ----
  const float tmain = thr[0];
  float zs = 0.f;
  for (int i = tid; i < V_TOK; i += TA) {
    float x = rowp[i];
    if (x >= tmain) zs += expf(x - m);
  }
  for (int o = 16; o; o >>= 1) zs += __shfl_xor(zs, o, 32);
  if (lane == 0) redF[wv] = zs;
  __syncthreads();
  if (wv == 0) {
    float t = redF[lane];
    for (int o = 16; o; o >>= 1) t += __shfl_xor(t, o, 32);
    if (lane == 0) s_bcastF = t;
  }
  __syncthreads();
  const float Z = s_bcastF;

  if (tid == 0) {
    float* o = stats + (size_t)r * 8;
    if (isExp) {
      // top-40 membership AND p >= 0.1*p_max  <=>  x >= max(t40, m + ln 0.1)
      float thr_out = fmaxf(thr[1], m + LN_ALPHA);
      o[0] = m; o[1] = Z; o[2] = thr_out;
    } else {
      o[4] = m; o[5] = Z; o[6] = thr[0];
    }
  }
}

// ================= Kernel B: streaming score output =================
__global__ void __launch_bounds__(256) score_kernel(
    const float* __restrict__ g_exp, const float* __restrict__ g_ama,
    const float* __restrict__ stats, float* __restrict__ out)
{
  const int r = blockIdx.y;
  const int c = blockIdx.x * 256 + threadIdx.x;
  if (c >= V_TOK) return;
  const float* st = stats + (size_t)r * 8;
  const float m_e = st[0], Z_e = st[1], thr = st[2];
  const float m_a = st[4], Z_a = st[5], t_a = st[6];
  const size_t idx = (size_t)r * V_TOK + c;

  float x = g_exp[idx];
  float res = neg_inf_f();
  if (x >= thr) {                       // <=40 lanes per row take this path
    float p = expf(x - m_e) / Z_e;
    float y = g_ama[idx];               // ama logit loaded only under mask
    float q = (y >= t_a) ? (expf(y - m_a) / Z_a) : 0.f;
    res = logf(p / (q + 1e-8f));
  }
  out[idx] = res;
}

// ================= launch =================
extern "C" void kernel_launch(void* const* d_in, const int* in_sizes, int n_in,
                              void* d_out, int out_size, void* d_ws, size_t ws_size,
                              hipStream_t stream) {
  const float* g_exp = (const float*)d_in[0];
  const float* g_ama = (const float*)d_in[1];
  float* out   = (float*)d_out;
  float* stats = (float*)d_ws;          // 512 rows * 8 floats = 16 KB

  // ~201 KB dynamic LDS: whole vocab row resident per workgroup (320 KB WGP LDS)
  // +4 floats of slack for the 16B phase-matching shift.
  size_t ldsBytes = (size_t)(V_TOK + 4) * sizeof(float);

  row_stats_kernel<<<dim3(2 * B_ROWS), dim3(TA), ldsBytes, stream>>>(g_exp, g_ama, stats);

  dim3 gridB((V_TOK + 255) / 256, B_ROWS);
  score_kernel<<<gridB, dim3(256), 0, stream>>>(g_exp, g_ama, stats, out);
}